// MambaBlock_10599979287086
// MI455X (gfx1250) — compile-verified
//
#include <hip/hip_runtime.h>
#include <stdint.h>

typedef __attribute__((ext_vector_type(2))) float v2f;
typedef __attribute__((ext_vector_type(8))) float v8f;

#define SSM_DT   0.001f
#define LN_EPS   1e-5f

// ---------------------------------------------------------------------------
// CDNA5 async global->LDS copy support (guarded so compile never breaks)
// ---------------------------------------------------------------------------
#ifdef __has_builtin
#if __has_builtin(__builtin_amdgcn_global_load_async_to_lds_b128)
#define USE_ASYNC_LDS 1
#endif
#endif
#ifndef USE_ASYNC_LDS
#define USE_ASYNC_LDS 0
#endif

#if USE_ASYNC_LDS
// Builtin signature (from hipcc diagnostic): param 0 is
// 'int __attribute__((vector_size(16))) addrspace(1)*', param 1 the LDS-side
// pointer, then two int immediates (offset, cpol).
typedef int v4i_g __attribute__((vector_size(16)));
typedef __attribute__((address_space(1))) v4i_g as1_v4i;
typedef __attribute__((address_space(3))) v4i_g as3_v4i;

// Generic LDS pointer's low 32 bits ARE the LDS byte offset (ISA aperture rule),
// so an int->as3-pointer round trip gives a valid LDS address without an
// addrspacecast.
__device__ __forceinline__ void async_cp16(const float* g, const float* lds) {
  __builtin_amdgcn_global_load_async_to_lds_b128(
      (as1_v4i*)(uintptr_t)g, (as3_v4i*)(uint32_t)(uintptr_t)lds, 0, 0);
}
__device__ __forceinline__ void wait_async0() {
#if __has_builtin(__builtin_amdgcn_s_wait_asynccnt)
  __builtin_amdgcn_s_wait_asynccnt(0);
#else
  asm volatile("s_wait_asynccnt 0" ::: "memory");
#endif
}
#endif

// ---------------------------------------------------------------------------
// Transpose (1024x1024 per batch): dst[b][i][j] = src[b][j][i]
// ---------------------------------------------------------------------------
__global__ __launch_bounds__(256) void transpose_bll(const float* __restrict__ src,
                                                     float* __restrict__ dst) {
  __shared__ float tile[32][33];
  int b = blockIdx.z;
  int i0 = blockIdx.y * 32, j0 = blockIdx.x * 32;
  int tx = threadIdx.x, ty = threadIdx.y;           // 32 x 8
  size_t base = (size_t)b << 20;                    // b * 1024 * 1024
#pragma unroll
  for (int k = 0; k < 32; k += 8)
    tile[ty + k][tx] = src[base + (size_t)(j0 + ty + k) * 1024 + i0 + tx];
  __syncthreads();
#pragma unroll
  for (int k = 0; k < 32; k += 8)
    dst[base + (size_t)(i0 + ty + k) * 1024 + j0 + tx] = tile[tx][ty + k];
}

// ---------------------------------------------------------------------------
// LayerNorm over last dim (C = 1024), one block per row, 256 threads
// ---------------------------------------------------------------------------
__global__ __launch_bounds__(256) void ln_kernel(const float* __restrict__ x,
                                                 const float* __restrict__ gamma,
                                                 const float* __restrict__ beta,
                                                 float* __restrict__ out, int C) {
  int row = blockIdx.x;
  int t = threadIdx.x;
  const float4 d = ((const float4*)(x + (size_t)row * C))[t];
  float s  = d.x + d.y + d.z + d.w;
  float ss = d.x * d.x + d.y * d.y + d.z * d.z + d.w * d.w;
  __shared__ float rs[256], rq[256];
  rs[t] = s; rq[t] = ss;
  __syncthreads();
  for (int off = 128; off > 0; off >>= 1) {
    if (t < off) { rs[t] += rs[t + off]; rq[t] += rq[t + off]; }
    __syncthreads();
  }
  float mean = rs[0] / (float)C;
  float var  = rq[0] / (float)C - mean * mean;
  float inv  = rsqrtf(var + LN_EPS);
  const float4 g = ((const float4*)gamma)[t];
  const float4 bb = ((const float4*)beta)[t];
  float4 o;
  o.x = (d.x - mean) * inv * g.x + bb.x;
  o.y = (d.y - mean) * inv * g.y + bb.y;
  o.z = (d.z - mean) * inv * g.z + bb.z;
  o.w = (d.w - mean) * inv * g.w + bb.w;
  ((float4*)(out + (size_t)row * C))[t] = o;
}

// ---------------------------------------------------------------------------
// A_bar = I + DT*A  (128x128)
// ---------------------------------------------------------------------------
__global__ void abar_kernel(const float* __restrict__ A, float* __restrict__ abar) {
  int i = blockIdx.x * blockDim.x + threadIdx.x;   // 16384
  int s = i >> 7, k = i & 127;
  abar[i] = (s == k ? 1.0f : 0.0f) + SSM_DT * A[i];
}

// ---------------------------------------------------------------------------
// WMMA GEMM: Cmat[M x N] = alpha * (A[M x K] @ Bw[N x K]^T)  (+ bias[N])
//   mode 0: plain           mode 1: += epA[row,col] * epB[col]
//   mode 2: += old Cmat value (residual accumulate)
// Block tile 128x64, BK=16, 8 waves of 32x32, fp32 WMMA 16x16x4.
// Double-buffered LDS; async DMA staging when available.
// ---------------------------------------------------------------------------
#define BM 128
#define BN 64
#define BK 16
#define LDA_S 20   // padded LDS row stride (floats): 16B-aligned, conflict-free

__global__ __launch_bounds__(256) void wmma_gemm(
    const float* __restrict__ A, const float* __restrict__ Bw,
    const float* __restrict__ bias, const float* __restrict__ epA,
    const float* __restrict__ epB, float* __restrict__ Cmat,
    int M, int N, int K, float alpha, int mode) {
  __shared__ __align__(16) float sA[2][BM * LDA_S];
  __shared__ __align__(16) float sB[2][BN * LDA_S];

  const int tid  = threadIdx.x;
  const int lane = tid & 31, wid = tid >> 5;
  const int hi = lane >> 4, lo = lane & 15;
  const int bm = blockIdx.y * BM, bn = blockIdx.x * BN;
  const int wm = (wid >> 1) * 32, wn = (wid & 1) * 32;

  v8f acc[2][2] = {};

  auto compute_tile = [&](const float* cA, const float* cB) {
#pragma unroll
    for (int kk = 0; kk < 4; kk++) {
      const int ko = kk * 4 + hi * 2;
      v2f aF[2], bF[2];
      aF[0] = *(const v2f*)(cA + (wm + lo) * LDA_S + ko);
      aF[1] = *(const v2f*)(cA + (wm + 16 + lo) * LDA_S + ko);
      bF[0] = *(const v2f*)(cB + (wn + lo) * LDA_S + ko);
      bF[1] = *(const v2f*)(cB + (wn + 16 + lo) * LDA_S + ko);
#pragma unroll
      for (int i = 0; i < 2; i++)
#pragma unroll
        for (int j = 0; j < 2; j++)
          acc[i][j] = __builtin_amdgcn_wmma_f32_16x16x4_f32(
              false, aF[i], false, bF[j], (short)0, acc[i][j], false, false);
    }
  };

#if USE_ASYNC_LDS
  // --- async DMA double-buffered pipeline ---
  auto stage = [&](int buf, int k0) {
#pragma unroll
    for (int j = 0; j < 2; j++) {           // A tile: 512 B128 slots, 2/thread
      int s = tid * 2 + j;
      int r = s >> 2, c4 = (s & 3) * 4;
      async_cp16(A + (size_t)(bm + r) * K + k0 + c4, &sA[buf][r * LDA_S + c4]);
    }
    {                                       // B tile: 256 B128 slots, 1/thread
      int r = tid >> 2, c4 = (tid & 3) * 4;
      async_cp16(Bw + (size_t)(bn + r) * K + k0 + c4, &sB[buf][r * LDA_S + c4]);
    }
  };

  stage(0, 0);
  wait_async0();
  __syncthreads();
  int cur = 0;
  for (int k0 = 0; k0 < K; k0 += BK) {
    if (k0 + BK < K) stage(cur ^ 1, k0 + BK);   // DMA next tile in background
    compute_tile(sA[cur], sB[cur]);             // 16 WMMAs on current tile
    wait_async0();
    __syncthreads();
    cur ^= 1;
  }
#else
  // --- fallback: synchronous single-buffer staging ---
  for (int k0 = 0; k0 < K; k0 += BK) {
#pragma unroll
    for (int j = 0; j < 2; j++) {
      int s = tid * 2 + j;
      int r = s >> 2, c4 = (s & 3) * 4;
      float4 v = *(const float4*)(A + (size_t)(bm + r) * K + k0 + c4);
      *(float4*)(&sA[0][r * LDA_S + c4]) = v;
    }
    {
      int r = tid >> 2, c4 = (tid & 3) * 4;
      float4 v = *(const float4*)(Bw + (size_t)(bn + r) * K + k0 + c4);
      *(float4*)(&sB[0][r * LDA_S + c4]) = v;
    }
    __syncthreads();
    compute_tile(sA[0], sB[0]);
    __syncthreads();
  }
#endif

  // Epilogue. C/D layout: VGPR r -> M = r + 8*hi, N = lo.
#pragma unroll
  for (int i = 0; i < 2; i++)
#pragma unroll
    for (int j = 0; j < 2; j++)
#pragma unroll
      for (int r = 0; r < 8; r++) {
        int row = bm + wm + i * 16 + r + hi * 8;
        int col = bn + wn + j * 16 + lo;
        size_t idx = (size_t)row * N + col;
        float v = alpha * acc[i][j][r];
        if (bias) v += bias[col];
        if (mode == 1)      v += epA[idx] * epB[col];
        else if (mode == 2) v += Cmat[idx];
        Cmat[idx] = v;
      }
}

// ---------------------------------------------------------------------------
// Sequential scan: h (16x128, rows 8..15 zero-padded) lives in LDS.
//   per step l: h = h @ A_bar^T + ub[:, l, :];  store hs[b*L + l][s] = h[b][s]
// One block, 8 waves; wave w owns state columns 16w..16w+15.
// A_bar fragments preloaded into registers (loop-invariant, 32 x float2).
// The next step's ub slab is async-DMA'd into LDS while WMMAs run.
// ---------------------------------------------------------------------------
#define SLDS 132   // 16 fragment rows -> 16 distinct LDS banks

__global__ __launch_bounds__(256) void scan_kernel(
    const float* __restrict__ ub, const float* __restrict__ abar,
    float* __restrict__ hs, int L) {
  __shared__ __align__(16) float hbuf[16 * SLDS];
#if USE_ASYNC_LDS
  __shared__ __align__(16) float ubs[2][8 * 128];
#endif
  const int tid = threadIdx.x;
  const int lane = tid & 31, wid = tid >> 5;
  const int hi = lane >> 4, lo = lane & 15;
  const int n0 = wid * 16;

  // Preload B fragments of A_bar (stored [s][k], K-major like the GEMM weights)
  v2f bF[32];
#pragma unroll
  for (int kk = 0; kk < 32; kk++)
    bF[kk] = *(const v2f*)(abar + (size_t)(n0 + lo) * 128 + kk * 4 + hi * 2);

#if USE_ASYNC_LDS
  auto prefetch_ub = [&](int buf, int l) {   // 8 rows x 128 floats, B128/thread
    int b = tid >> 5, s4 = (tid & 31) * 4;
    async_cp16(ub + ((size_t)b * L + l) * 128 + s4, &ubs[buf][b * 128 + s4]);
  };
  prefetch_ub(0, 0);
#endif

  for (int i = tid; i < 16 * SLDS; i += 256) hbuf[i] = 0.0f;
#if USE_ASYNC_LDS
  wait_async0();
#endif
  __syncthreads();

  for (int l = 0; l < L; l++) {
#if USE_ASYNC_LDS
    if (l + 1 < L) prefetch_ub((l + 1) & 1, l + 1);  // hide global latency
#endif
    v8f c = {};
#pragma unroll
    for (int kk = 0; kk < 32; kk++) {
      v2f a = *(const v2f*)(hbuf + lo * SLDS + kk * 4 + hi * 2);
      c = __builtin_amdgcn_wmma_f32_16x16x4_f32(
          false, a, false, bF[kk], (short)0, c, false, false);
    }
    __syncthreads();            // everyone done reading h before overwrite
#pragma unroll
    for (int r = 0; r < 8; r++) {
      int m = r + hi * 8;       // padded batch row
      int n = n0 + lo;          // state column
      float v = c[r];
      if (m < 8) {
#if USE_ASYNC_LDS
        v += ubs[l & 1][m * 128 + n];
#else
        v += ub[((size_t)m * L + l) * 128 + n];
#endif
        hs[((size_t)m * L + l) * 128 + n] = v;
      }
      hbuf[m * SLDS + n] = v;
    }
#if USE_ASYNC_LDS
    wait_async0();
#endif
    __syncthreads();
  }
}

// ---------------------------------------------------------------------------
// Launcher
// ---------------------------------------------------------------------------
extern "C" void kernel_launch(void* const* d_in, const int* in_sizes, int n_in,
                              void* d_out, int out_size, void* d_ws, size_t ws_size,
                              hipStream_t stream) {
  (void)in_sizes; (void)n_in; (void)out_size; (void)ws_size;
  const float* x     = (const float*)d_in[0];
  const float* Aall  = (const float*)d_in[1];
  const float* Bp    = (const float*)d_in[2];
  const float* Cp    = (const float*)d_in[3];
  const float* Dp    = (const float*)d_in[4];
  const float* Wi    = (const float*)d_in[5];
  const float* bi    = (const float*)d_in[6];
  const float* Wo    = (const float*)d_in[7];
  const float* bo    = (const float*)d_in[8];
  const float* gamma = (const float*)d_in[9];
  const float* beta  = (const float*)d_in[10];
  float* out = (float*)d_out;

  const int Bn = 8, L = 1024, C = 1024, S = 128, depth = 4;
  const size_t nBLC = (size_t)Bn * L * C;    // 8Mi floats
  const size_t nBLS = (size_t)Bn * L * S;    // 1Mi floats

  float* seq  = (float*)d_ws;                // (B, L, C)
  float* nbuf = seq  + nBLC;                 // LN output, reused as y buffer
  float* ubuf = nbuf + nBLC;                 // u
  float* ubb  = ubuf + nBLC;                 // DT * u @ Bp^T   (b*L+l, S)
  float* hsb  = ubb  + nBLS;                 // scan states     (b*L+l, S)
  float* abar = hsb  + nBLS;                 // I + DT*A (128x128)

  dim3 tb(32, 8);
  dim3 tg(32, 32, Bn);
  transpose_bll<<<tg, tb, 0, stream>>>(x, seq);   // seq[b][l][c] = x[b][c][l]

  for (int i = 0; i < depth; i++) {
    ln_kernel<<<Bn * L, 256, 0, stream>>>(seq, gamma + i * C, beta + i * C, nbuf, C);

    // u = n @ Wi^T + bi
    wmma_gemm<<<dim3(C / BN, (Bn * L) / BM), 256, 0, stream>>>(
        nbuf, Wi + (size_t)i * C * C, bi + i * C, nullptr, nullptr,
        ubuf, Bn * L, C, C, 1.0f, 0);

    // ub = DT * (u @ Bp^T)
    wmma_gemm<<<dim3(S / BN, (Bn * L) / BM), 256, 0, stream>>>(
        ubuf, Bp + (size_t)i * S * C, nullptr, nullptr, nullptr,
        ubb, Bn * L, S, C, SSM_DT, 0);

    abar_kernel<<<(S * S) / 256, 256, 0, stream>>>(Aall + (size_t)i * S * S, abar);

    scan_kernel<<<1, 256, 0, stream>>>(ubb, abar, hsb, L);

    // y = hs @ Cp^T + u * Dp   (into nbuf)
    wmma_gemm<<<dim3(C / BN, (Bn * L) / BM), 256, 0, stream>>>(
        hsb, Cp + (size_t)i * C * S, nullptr, ubuf, Dp + i * C,
        nbuf, Bn * L, C, S, 1.0f, 1);

    // seq += y @ Wo^T + bo
    wmma_gemm<<<dim3(C / BN, (Bn * L) / BM), 256, 0, stream>>>(
        nbuf, Wo + (size_t)i * C * C, bo + i * C, nullptr, nullptr,
        seq, Bn * L, C, C, 1.0f, 2);
  }

  transpose_bll<<<tg, tb, 0, stream>>>(seq, out); // out[b][c][l] = seq[b][l][c]
}